// SwitchboxElement_74500502716935
// MI455X (gfx1250) — compile-verified
//
#include <hip/hip_runtime.h>
#include <stdint.h>

typedef float v2f  __attribute__((ext_vector_type(2)));
typedef float v8f  __attribute__((ext_vector_type(8)));
typedef unsigned int u32x4 __attribute__((ext_vector_type(4)));
typedef int   i32x4 __attribute__((ext_vector_type(4)));
typedef int   i32x8 __attribute__((ext_vector_type(8)));

#define N_FULL      (1u << 25)          // tree width (leaves)
#define NX          (N_FULL - 1u)       // length of x
#define BLK_ELEMS   16384u              // per block  -> levels 0..13
#define WAVE_ELEMS  2048u               // per wave   (TDM tile)
#define PAD_STRIDE  66                  // DWORDs per 64-elem lane chunk (64 data + 2 TDM pad)
#define WAVE_LDS_DW (32 * PAD_STRIDE)   // 2112 DWORDs per wave slice
#define NBLK        2048                // N_FULL / BLK_ELEMS

__device__ __forceinline__ int rfl(unsigned v) {
  return __builtin_amdgcn_readfirstlane((int)v);
}

// Reduce 8 values (r8[0..7] in LDS) through 3 tree levels using one
// V_WMMA_F32_16X16X4_F32.  Coefficient for r8[i] factors per-bit:
//   coef(i) = f0(bit0 i) * f1(bit1 i) * f2(bit2 i)
// Place r0..r3 in A row 0, r4..r7 in A row 1 (16x4 f32 A layout:
// lanes 0-15: VGPR0=K0, VGPR1=K1; lanes 16-31: VGPR0=K2, VGPR1=K3).
// B[k][n] = f0(bit0 k)*f1(bit1 k) broadcast over n (B 4x16: VGPR0 rows 0/1
// in lane halves, VGPR1 rows 2/3).  Then out = fa2*D[0][0] + fb2*D[1][0].
__device__ __forceinline__ float wmma_reduce8(const float* r8, int lane,
    float fa0, float fb0, float fa1, float fb1, float fa2, float fb2)
{
  int li  = lane & 15;
  int act = (li < 2) ? 1 : 0;
  int idx = li * 4 + ((lane >= 16) ? 2 : 0);
  idx = act ? idx : 0;
  float m  = (float)act;
  float a0 = r8[idx] * m;        // A[row][k]   for this lane
  float a1 = r8[idx + 1] * m;    // A[row][k+1]
  float k00 = fa0 * fa1, k01 = fb0 * fa1, k10 = fa0 * fb1, k11 = fb0 * fb1;
  float b0 = (lane < 16) ? k00 : k01;   // B rows 0/1
  float b1 = (lane < 16) ? k10 : k11;   // B rows 2/3
  v8f c = {0.f, 0.f, 0.f, 0.f, 0.f, 0.f, 0.f, 0.f};
  v2f A; A.x = a0; A.y = a1;
  v2f B; B.x = b0; B.y = b1;
  c = __builtin_amdgcn_wmma_f32_16x16x4_f32(false, A, false, B, (short)0, c,
                                            false, false);
  // lane 0: c[0] = D[0][0], c[1] = D[1][0]
  return fa2 * c[0] + fb2 * c[1];
}

// Kernel 1: levels 0..13.  Each block owns 16384 contiguous leaves,
// each wave TDM-loads its 2048-leaf slice into LDS with 2-DWORD padding
// every 64 DWORDs (conflict-free per-lane chunks), each thread folds its
// 64 leaves in registers, wave folds 32->1 via shfl_xor, block folds 8->1
// via WMMA.
__global__ __launch_bounds__(256) void k1(const float* __restrict__ x,
                                          const float* __restrict__ cw,
                                          const float* __restrict__ s,
                                          float* __restrict__ partials)
{
  __shared__ float stage[8 * WAVE_LDS_DW];   // 67584 B
  __shared__ float rbuf[8];
  const int tid  = threadIdx.x;
  const int w    = tid >> 5;
  const int lane = tid & 31;

  float fa[14], fb[14];
#pragma unroll
  for (int L = 0; L < 14; ++L) {
    float sv = s[L];
    fa[L] = 0.5f * (1.0f + sv);
    fb[L] = 0.5f * (1.0f - sv);
  }
  const float cw0 = cw[0];

  const unsigned start     = (unsigned)blockIdx.x * BLK_ELEMS + (unsigned)w * WAVE_ELEMS;
  const unsigned remaining = NX - start;   // always >= 2047; OOB tail reads return 0
  const unsigned lds_base  = (unsigned)(uintptr_t)(&stage[0]) + (unsigned)w * (WAVE_LDS_DW * 4u);
  const unsigned long long ga = (unsigned long long)(uintptr_t)x
                              + (unsigned long long)start * 4ull;

  // ---- TDM descriptor (1D tile, data_size=4B, pad 2 DW every 64 DW) ----
  u32x4 g0;
  g0[0] = (unsigned)rfl(1u);                                   // count=1
  g0[1] = (unsigned)rfl(lds_base);                             // lds_addr
  g0[2] = (unsigned)rfl((unsigned)(ga & 0xFFFFFFFFull));       // global lo
  g0[3] = (unsigned)rfl(((unsigned)(ga >> 32) & 0x01FFFFFFu) | (2u << 30)); // hi | type=2
  i32x8 g1;
  g1[0] = rfl((2u << 16) | (1u << 20) | (5u << 22) | (1u << 25)); // dsz=4B,pad_en,int=64DW,amt=2DW
  g1[1] = rfl((remaining & 0xFFFFu) << 16);                    // tensor_dim0 lo16
  g1[2] = rfl((remaining >> 16) | (1u << 16));                 // tensor_dim0 hi16 | tensor_dim1=1
  g1[3] = rfl(WAVE_ELEMS << 16);                               // tile_dim0=2048
  g1[4] = 0;                                                   // tile_dim1/2 unused
  g1[5] = rfl(remaining);                                      // dim0 stride (unused for 1D)
  g1[6] = 0;
  g1[7] = 0;
  i32x4 gz4 = {0, 0, 0, 0};
#if defined(__clang_major__) && __clang_major__ >= 23
  i32x8 gz8 = {0, 0, 0, 0, 0, 0, 0, 0};
  __builtin_amdgcn_tensor_load_to_lds(g0, g1, gz4, gz4, gz8, 0);
#else
  __builtin_amdgcn_tensor_load_to_lds(g0, g1, gz4, gz4, 0);
#endif
  __builtin_amdgcn_s_wait_tensorcnt(0);
  __asm__ volatile("" ::: "memory");

  // ---- levels 0..5 in registers (64 contiguous leaves per thread) ----
  const float2* src = (const float2*)&stage[w * WAVE_LDS_DW + lane * PAD_STRIDE];
  const bool lastThread = (start + (unsigned)lane * 64u + 63u) == (N_FULL - 1u);
  float u[32];
#pragma unroll
  for (int k = 0; k < 32; ++k) {
    float2 p = src[k];
    if (k == 31 && lastThread) p.y = cw0;  // final leaf is the trained constant
    u[k] = fa[0] * p.x + fb[0] * p.y;
  }
#pragma unroll
  for (int L = 1; L < 6; ++L) {
#pragma unroll
    for (int j = 0; j < (32 >> L); ++j)
      u[j] = fa[L] * u[2 * j] + fb[L] * u[2 * j + 1];
  }
  float val = u[0];

  // ---- levels 6..10 across the wave (wave32 butterflies) ----
#pragma unroll
  for (int m = 0; m < 5; ++m) {
    float o = __shfl_xor(val, 1 << m, 32);
    int hi = (lane >> m) & 1;
    float lo_v = hi ? o : val;
    float hi_v = hi ? val : o;
    val = fa[6 + m] * lo_v + fb[6 + m] * hi_v;
  }

  // ---- levels 11..13 across 8 waves via WMMA ----
  if (lane == 0) rbuf[w] = val;
  __syncthreads();
  float out = wmma_reduce8(rbuf, lane, fa[11], fb[11], fa[12], fb[12], fa[13], fb[13]);
  if (tid == 0) partials[blockIdx.x] = out;
}

// Kernel 2: levels 14..24 over the 2048 block partials (one block).
__global__ __launch_bounds__(256) void k2(const float* __restrict__ p,
                                          const float* __restrict__ s,
                                          float* __restrict__ out)
{
  __shared__ float rbuf[8];
  const int tid  = threadIdx.x;
  const int w    = tid >> 5;
  const int lane = tid & 31;

  float fa[11], fb[11];
#pragma unroll
  for (int L = 0; L < 11; ++L) {
    float sv = s[14 + L];
    fa[L] = 0.5f * (1.0f + sv);
    fb[L] = 0.5f * (1.0f - sv);
  }

  const float4* p4 = (const float4*)(p + tid * 8);
  float4 q0 = p4[0], q1 = p4[1];
  float u[8] = {q0.x, q0.y, q0.z, q0.w, q1.x, q1.y, q1.z, q1.w};
#pragma unroll
  for (int L = 0; L < 3; ++L) {            // levels 14..16
#pragma unroll
    for (int j = 0; j < (4 >> L); ++j)
      u[j] = fa[L] * u[2 * j] + fb[L] * u[2 * j + 1];
  }
  float val = u[0];
#pragma unroll
  for (int m = 0; m < 5; ++m) {            // levels 17..21
    float o = __shfl_xor(val, 1 << m, 32);
    int hi = (lane >> m) & 1;
    float lo_v = hi ? o : val;
    float hi_v = hi ? val : o;
    val = fa[3 + m] * lo_v + fb[3 + m] * hi_v;
  }
  if (lane == 0) rbuf[w] = val;
  __syncthreads();
  float res = wmma_reduce8(rbuf, lane, fa[8], fb[8], fa[9], fb[9], fa[10], fb[10]); // 22..24
  if (tid == 0) out[0] = res;
}

extern "C" void kernel_launch(void* const* d_in, const int* in_sizes, int n_in,
                              void* d_out, int out_size, void* d_ws, size_t ws_size,
                              hipStream_t stream) {
  const float* x  = (const float*)d_in[0];   // 2^25 - 1 floats
  const float* cw = (const float*)d_in[1];   // 1 float
  const float* s  = (const float*)d_in[2];   // 25 floats
  float* partials = (float*)d_ws;            // 2048 floats
  k1<<<NBLK, 256, 0, stream>>>(x, cw, s, partials);
  k2<<<1, 256, 0, stream>>>(partials, s, (float*)d_out);
}